// ConvLSTMCell_8151847928118
// MI455X (gfx1250) — compile-verified
//
#include <hip/hip_runtime.h>

typedef __attribute__((ext_vector_type(2))) float v2f;
typedef __attribute__((ext_vector_type(8))) float v8f;

#define NB   4      // batch
#define HID  64     // hidden channels
#define OC   256    // 4*HID gate channels
#define TT   16     // timesteps
#define HS   64     // spatial H
#define WS   64     // spatial W
#define KTOT 576    // K = HID * 9
#define NQ   144    // K / 4 : wmma chunks
#define NPX  32     // pixels per block tile (row segment)
#define PCOL 34     // patch cols = NPX + 2 halo
#define PSTR 35     // padded patch row stride
#define PCH  (3*PSTR)
#define ASTR 33     // im2col pair-row stride in float2 (NPX + 1 pad)
#define TSTR 257    // gate-tile stride (256 + 1 pad)

#define APC_BYTES   (2*NQ*ASTR*8)                 // 76032 B (persistent)
#define PATCH_BYTES (HID*PCH*4)                   // 26880 B (phase 1)
#define GTILE_BYTES (NPX*TSTR*4)                  // 32896 B (phase 3, reuses patch)
#define SMEM_BYTES  (APC_BYTES + GTILE_BYTES)     // 108928 B

__device__ __forceinline__ float sigm(float x)  { return 1.0f / (1.0f + expf(-x)); }
__device__ __forceinline__ float lrelu(float x) { return x > 0.0f ? x : 0.01f * x; }

// ---------------------------------------------------------------------------
// One-time weight repack: Wp[pairRow ph][o] = {W[o][2ph], W[o][2ph+1]}
// -> B-fragment load for k-chunk q, lane-half h is Wp[(2q+h)*OC + n], fully
//    coalesced (two runs of 16 consecutive float2 per wave load).
// ---------------------------------------------------------------------------
__global__ __launch_bounds__(256) void pack_w(const float* __restrict__ Wt,
                                              float* __restrict__ Wp) {
    int idx = blockIdx.x * 256 + threadIdx.x;    // over 2*NQ*OC = 73728
    if (idx >= 2 * NQ * OC) return;
    int o  = idx & (OC - 1);
    int ph = idx >> 8;                           // pair row, k = 2*ph
    Wp[2 * idx]     = Wt[o * KTOT + 2 * ph];
    Wp[2 * idx + 1] = Wt[o * KTOT + 2 * ph + 1];
}

// ---------------------------------------------------------------------------
// t = 0: gates directly on xi[:, :, 0]
// ---------------------------------------------------------------------------
__global__ __launch_bounds__(256) void lstm_t0(const float* __restrict__ xi,
                                               float* __restrict__ out,
                                               float* __restrict__ cstate) {
    unsigned idx = blockIdx.x * 256u + threadIdx.x;      // NB*HID*HS*WS = 1M
    if (idx >= (unsigned)NB * HID * HS * WS) return;
    int sp = idx & (HS * WS - 1);
    int ch = (idx >> 12) & (HID - 1);
    int bb = idx >> 18;

    size_t xibase = ((size_t)bb * OC) * TT * HS * WS + sp;
    const size_t cs = (size_t)TT * HS * WS;
    float iv = sigm (xi[xibase + (size_t)(ch          ) * cs]);
    float ov = sigm (xi[xibase + (size_t)(ch + 2 * HID) * cs]);
    float gv = tanhf(xi[xibase + (size_t)(ch + 3 * HID) * cs]);

    float c0 = iv * gv;
    cstate[(size_t)(bb * HID + ch) * HS * WS + sp] = c0;
    out[((size_t)(bb * HID + ch) * TT + 0) * HS * WS + sp] = ov * lrelu(c0);
}

// ---------------------------------------------------------------------------
// t >= 1: block = 512 threads (16 waves), tile = 32 pixels x 256 channels.
// Wave w -> out-channels [16w,16w+16), two 16-pixel subtiles (2 accumulators).
// Steady state per k-chunk: 1 coalesced global b64 (B) + 2 ds b64 (A)
// + 2 x V_WMMA_F32_16X16X4_F32.
// ---------------------------------------------------------------------------
__global__ __launch_bounds__(512) void conv_lstm_step(
        const float* __restrict__ xi, const v2f* __restrict__ Wp,
        const float* __restrict__ bias, float* __restrict__ out,
        float* __restrict__ cstate, int t) {
    __shared__ __align__(16) char smem[SMEM_BYTES];
    v2f*   apc   = (v2f*)smem;                   // [2*NQ][ASTR] pair buffer
    float* patch = (float*)(smem + APC_BYTES);   // phase 1 only
    float* gtile = (float*)(smem + APC_BYTES);   // phase 3, reuses patch space

    const int tid  = threadIdx.x;
    const int lane = tid & 31;
    const int wv   = tid >> 5;                   // 0..15
    const int x0   = blockIdx.x * NPX;           // 0 or 32
    const int y    = blockIdx.y;
    const int bb   = blockIdx.z;

    // ---- phase 1: stage zero-padded h_{t-1} halo (64ch x 3 x 34) ----------
    for (int e = tid; e < HID * 3 * PCOL; e += 512) {
        int i = e / (3 * PCOL);
        int r = (e % (3 * PCOL)) / PCOL;
        int c = e % PCOL;
        int yy = y + r - 1;
        int xx = x0 + c - 1;
        float v = 0.0f;
        if (yy >= 0 && yy < HS && xx >= 0 && xx < WS)
            v = out[((size_t)(bb * HID + i) * TT + (t - 1)) * HS * WS + yy * WS + xx];
        patch[i * PCH + r * PSTR + c] = v;
    }
    __syncthreads();

    // ---- phase 2a: im2col pair buffer, shared by all 16 waves -------------
    // apc[pj][px] = { A[px][2pj], A[px][2pj+1] },  k = i*9 + r*3 + c
    for (int e = tid; e < 2 * NQ * NPX; e += 512) {
        int pj = e >> 5;
        int px = e & (NPX - 1);
        int k0 = 2 * pj, k1 = k0 + 1;
        int i0 = k0 / 9, r0 = (k0 % 9) / 3, c0 = k0 % 3;
        int i1 = k1 / 9, r1 = (k1 % 9) / 3, c1 = k1 % 3;
        v2f v;
        v.x = patch[i0 * PCH + r0 * PSTR + px + c0];
        v.y = patch[i1 * PCH + r1 * PSTR + px + c1];
        apc[pj * ASTR + px] = v;
    }
    __syncthreads();   // patch dead from here; gtile may reuse its space

    // ---- phase 2b: GEMM over K = 576 --------------------------------------
    const int n0   = wv * 16;
    const int m    = lane & 15;
    const int half = lane >> 4;                  // K split across lane halves
    const v2f* bptr = Wp + (size_t)half * OC + (n0 + m);
    const v2f* aptr = apc + half * ASTR + m;

    v8f acc0 = {}, acc1 = {};
#pragma unroll 4
    for (int q = 0; q < NQ; ++q) {
        v2f b  = bptr[0];                        // coalesced, L2-hot
        v2f a0 = aptr[0];                        // subtile 0 (pixels 0..15)
        v2f a1 = aptr[16];                       // subtile 1 (pixels 16..31)
        // WGP-scope prefetch: pull upcoming W stream into near cache
        __builtin_prefetch((const void*)(bptr + 16 * OC), 0, 3);
        acc0 = __builtin_amdgcn_wmma_f32_16x16x4_f32(false, a0, false, b,
                                                     (short)0, acc0, false, false);
        acc1 = __builtin_amdgcn_wmma_f32_16x16x4_f32(false, a1, false, b,
                                                     (short)0, acc1, false, false);
        bptr += 2 * OC;
        aptr += 2 * ASTR;
    }

    // ---- phase 3: scatter D + bias + xi[:,:,t] into gate tile -------------
    // D layout: VGPR j -> pixel = (lane<16?0:8)+j (+16 for acc1), chan = n0+(lane&15)
    {
        const int nch   = n0 + m;
        const int pbase = half * 8;
        const float bv  = bias[nch];
        const float* xirow =
            xi + (((size_t)bb * OC + nch) * TT + t) * HS * WS + y * WS + x0;
#pragma unroll
        for (int j = 0; j < 8; ++j) {
            int p = pbase + j;
            gtile[p * TSTR + nch]        = acc0[j] + bv + xirow[p];
            gtile[(p + 16) * TSTR + nch] = acc1[j] + bv + xirow[p + 16];
        }
    }
    __syncthreads();

    // ---- phase 4: gates, c/h update (32px x 64ch, 4 per thread) -----------
    for (int e = tid; e < NPX * HID; e += 512) {
        int p  = e >> 6;
        int ch = e & (HID - 1);
        float iv = sigm (gtile[p * TSTR + ch]);
        float fv = sigm (gtile[p * TSTR + HID + ch]);
        float ov = sigm (gtile[p * TSTR + 2 * HID + ch]);
        float gv = tanhf(gtile[p * TSTR + 3 * HID + ch]);

        size_t cidx = (size_t)(bb * HID + ch) * HS * WS + y * WS + x0 + p;
        float cn = fv * cstate[cidx] + iv * gv;
        cstate[cidx] = cn;
        out[((size_t)(bb * HID + ch) * TT + t) * HS * WS + y * WS + x0 + p] =
            ov * lrelu(cn);
    }
}

// ---------------------------------------------------------------------------
extern "C" void kernel_launch(void* const* d_in, const int* in_sizes, int n_in,
                              void* d_out, int out_size, void* d_ws, size_t ws_size,
                              hipStream_t stream) {
    const float* xi   = (const float*)d_in[0];   // (4, 256, 16, 64, 64)
    const float* Wt   = (const float*)d_in[1];   // (256, 64, 3, 3)
    const float* bias = (const float*)d_in[2];   // (256,)
    float* out    = (float*)d_out;               // (4, 64, 16, 64, 64)
    float* cstate = (float*)d_ws;                // 4 MB
    float* WpRaw  = (float*)((char*)d_ws + (size_t)NB * HID * HS * WS * sizeof(float));

    pack_w<<<(2 * NQ * OC + 255) / 256, 256, 0, stream>>>(Wt, WpRaw);
    lstm_t0<<<(NB * HID * HS * WS + 255) / 256, 256, 0, stream>>>(xi, out, cstate);

    dim3 grid(WS / NPX, HS, NB);
    for (int t = 1; t < TT; ++t)
        conv_lstm_step<<<grid, 512, 0, stream>>>(xi, (const v2f*)WpRaw, bias,
                                                 out, cstate, t);
}